// RandLANet_47983374631064
// MI455X (gfx1250) — compile-verified
//
#include <hip/hip_runtime.h>
#include <hip/hip_bf16.h>
#include <stddef.h>

// ---------------------------------------------------------------------------
// RandLANet block on MI455X (gfx1250, wave32).
// f32 WMMA (V_WMMA_F32_16X16X4_F32) drives:
//   * KNN gram (pairwise dot products, 16x16 tiles, K=4 with xyz0 padding)
//   * LSE encoder MLP (enc[16x10] @ W^T, padded to K=12)
//   * attentive-pooling score matmul (x[16x64] @ Ws^T)
// KNN candidate staging uses the gfx1250 async global->LDS path
// (GLOBAL_LOAD_ASYNC_TO_LDS_B32, ASYNCcnt + s_wait_asynccnt).
// BatchNorm training-mode stats via LDS partials + global f32 atomics.
// ---------------------------------------------------------------------------

typedef float v2f __attribute__((ext_vector_type(2)));
typedef float v8f __attribute__((ext_vector_type(8)));

#define BB 4
#define NN 8192
#define KK 16
#define D_IN 32
#define D_OUT 64
#define HH 32

// stats region layout (floats): per group [sum C][sumsq C][mean C][invstd C]
#define GRP_LSE1 0      // C=32  -> 128 floats
#define GRP_P1   128    // C=32
#define GRP_LSE2 256    // C=32
#define GRP_P2   384    // C=64  -> 256 floats
#define GRP_SC   640    // C=128 -> 512 floats
#define STATS_FLOATS 1152

__device__ inline v2f mkv2(float x, float y) { v2f r; r.x = x; r.y = y; return r; }
__device__ inline v8f zero8() { v8f z = {0.f,0.f,0.f,0.f,0.f,0.f,0.f,0.f}; return z; }

__device__ inline v8f wmma4(v2f a, v2f b, v8f c) {
#if __has_builtin(__builtin_amdgcn_wmma_f32_16x16x4_f32)
    // (neg_a, A, neg_b, B, c_mod, C, reuse_a, reuse_b)
    return __builtin_amdgcn_wmma_f32_16x16x4_f32(false, a, false, b, (short)0, c,
                                                 false, false);
#else
    #pragma unroll
    for (int i = 0; i < 8; ++i) c[i] += a.x * b.x + a.y * b.y;
    return c;
#endif
}

#if defined(__gfx1250__) && __has_builtin(__builtin_amdgcn_global_load_async_to_lds_b32)
#define HAVE_ASYNC_LDS 1
#endif

#ifdef HAVE_ASYNC_LDS
#define AS1 __attribute__((address_space(1)))
#define AS3 __attribute__((address_space(3)))
__device__ inline void async_copy_b32(float* lds, const float* g) {
    __builtin_amdgcn_global_load_async_to_lds_b32(
        (AS1 int*)g, (AS3 int*)lds, /*offset=*/0, /*cpol=*/0);
}
__device__ inline void wait_asynccnt0() {
#if __has_builtin(__builtin_amdgcn_s_wait_asynccnt)
    __builtin_amdgcn_s_wait_asynccnt(0);
#else
    asm volatile("s_wait_asynccnt 0x0" ::: "memory");
#endif
}
#endif

// ---------------------------------------------------------------------------
// Kernel 1: brute-force KNN (self included), idx ordered by ascending d2.
// One wave handles 16 queries; block = 8 waves = 128 queries of one batch.
// A-frag = candidate tile (16x4: xyz0) pre-swizzled in LDS at staging time,
// B-frag = query tile (4x16) in registers.
// D[m][n] = cand_m . query_n ; d2 = |q|^2 + |c|^2 - 2*dot.
// Streaming top-16 per lane kept in LDS (kth cached in a VGPR);
// lane L / L+16 lists merged at the end.
// ---------------------------------------------------------------------------
__global__ __launch_bounds__(256) void knn_kernel(const float* __restrict__ pos,
                                                  int* __restrict__ idx_out) {
    __shared__ v2f   afrag[128][2];   // [cand][half]: {x,y} then {z,0}
    __shared__ float cnorm[128];
#ifdef HAVE_ASYNC_LDS
    __shared__ float staging[384];    // raw xyz chunk landed by async DMA
#endif
    __shared__ float topd[8][32][16];
    __shared__ int   topi[8][32][16];

    const int w    = threadIdx.x >> 5;
    const int lane = threadIdx.x & 31;
    const int b    = blockIdx.x >> 6;               // 64 blocks per batch
    const int q0   = (blockIdx.x & 63) * 128 + w * 16;
    const int m    = lane & 15;
    const int half = lane >> 4;

    const float* posb = pos + (size_t)b * NN * 3;
    const float* qp   = posb + (size_t)(q0 + m) * 3;
    const float qx = qp[0], qy = qp[1], qz = qp[2];
    const float qn = qx * qx + qy * qy + qz * qz;
    const v2f   bq = half ? mkv2(qz, 0.f) : mkv2(qx, qy);

    float* td = topd[w][lane];
    int*   ti = topi[w][lane];
    #pragma unroll
    for (int k = 0; k < 16; ++k) { td[k] = 3.0e38f; ti[k] = 0x7fffffff; }
    float kth = 3.0e38f;              // register-cached td[15]

    for (int t0 = 0; t0 < NN; t0 += 128) {
        __syncthreads();
        const float* gsrc = posb + (size_t)t0 * 3;  // 384 dwords
#ifdef HAVE_ASYNC_LDS
        if (threadIdx.x < 128) {
            async_copy_b32(&staging[threadIdx.x      ], gsrc + threadIdx.x);
            async_copy_b32(&staging[threadIdx.x + 128], gsrc + threadIdx.x + 128);
            async_copy_b32(&staging[threadIdx.x + 256], gsrc + threadIdx.x + 256);
        }
        wait_asynccnt0();
        __syncthreads();
        if (threadIdx.x < 128) {
            const float x = staging[3 * threadIdx.x + 0];
            const float y = staging[3 * threadIdx.x + 1];
            const float z = staging[3 * threadIdx.x + 2];
            afrag[threadIdx.x][0] = mkv2(x, y);
            afrag[threadIdx.x][1] = mkv2(z, 0.f);
            cnorm[threadIdx.x] = x * x + y * y + z * z;
        }
#else
        if (threadIdx.x < 128) {
            const float* cp = gsrc + (size_t)threadIdx.x * 3;
            const float x = cp[0], y = cp[1], z = cp[2];
            afrag[threadIdx.x][0] = mkv2(x, y);
            afrag[threadIdx.x][1] = mkv2(z, 0.f);
            cnorm[threadIdx.x] = x * x + y * y + z * z;
        }
#endif
        if (t0 + 128 < NN && threadIdx.x < 128)
            __builtin_prefetch(posb + (size_t)(t0 + 128 + threadIdx.x) * 3, 0, 0);
        __syncthreads();

        for (int s = 0; s < 8; ++s) {
            const v2f a = afrag[s * 16 + m][half];   // uniform-shape ds_load_b64
            v8f d = zero8();
            d = wmma4(a, bq, d);
            const float4* c4 = (const float4*)&cnorm[s * 16 + half * 8];
            const float4 n0 = c4[0], n1 = c4[1];
            const float cn[8] = {n0.x, n0.y, n0.z, n0.w, n1.x, n1.y, n1.z, n1.w};
            #pragma unroll
            for (int v = 0; v < 8; ++v) {
                const float dist = qn + cn[v] - 2.f * d[v];
                const int   ci   = t0 + s * 16 + half * 8 + v;
                if (dist < kth) {
                    int j = 15;
                    #pragma unroll 1
                    while (j > 0 && td[j - 1] > dist) {
                        td[j] = td[j - 1]; ti[j] = ti[j - 1]; --j;
                    }
                    td[j] = dist; ti[j] = ci;
                    kth = td[15];
                }
            }
        }
    }
    __syncthreads();
    if (lane < 16) {
        const float* da = topd[w][lane];      const int* ia = topi[w][lane];
        const float* db = topd[w][lane + 16]; const int* ib = topi[w][lane + 16];
        int pa = 0, pb2 = 0;
        int* out = idx_out + ((size_t)b * NN + (q0 + lane)) * KK;
        for (int k = 0; k < 16; ++k) {
            const bool ta = (pb2 >= 16) ||
                            (pa < 16 && (da[pa] < db[pb2] ||
                                         (da[pa] == db[pb2] && ia[pa] < ib[pb2])));
            if (ta) out[k] = ia[pa++]; else out[k] = ib[pb2++];
        }
    }
}

// ---------------------------------------------------------------------------
// enc row m = [cx,cy,cz, nx,ny,nz, rx,ry,rz, d2]; A-frags over 3 K=4 steps.
// ---------------------------------------------------------------------------
struct EncFrags { v2f a[3]; };

__device__ inline EncFrags build_enc_frags(const float* __restrict__ pos,
                                           const int* __restrict__ idxp,
                                           int b, int n, int lane) {
    const int m = lane & 15, half = lane >> 4;
    const float* cp = pos + ((size_t)b * NN + n) * 3;
    const float cx = cp[0], cy = cp[1], cz = cp[2];
    const int j = idxp[m];
    const float* np_ = pos + ((size_t)b * NN + j) * 3;
    const float nx = np_[0], ny = np_[1], nz = np_[2];
    const float rx = cx - nx, ry = cy - ny, rz = cz - nz;
    const float d2 = rx * rx + ry * ry + rz * rz;
    EncFrags f;
    if (half == 0) { f.a[0] = mkv2(cx, cy); f.a[1] = mkv2(ny, nz); f.a[2] = mkv2(rz, d2); }
    else           { f.a[0] = mkv2(cz, nx); f.a[1] = mkv2(rx, ry); f.a[2] = mkv2(0.f, 0.f); }
    return f;
}

// B-frags for lse weight W (32,10) row-major: B[k][d] = W[d][k], padded K 10->12.
__device__ inline void lse_wfrags(const float* __restrict__ w, int lane, v2f bw[2][3]) {
    const int d0 = lane & 15, half = lane >> 4;
    #pragma unroll
    for (int t = 0; t < 2; ++t) {
        const float* wr = w + (size_t)(t * 16 + d0) * 10;
        if (half == 0) { bw[t][0] = mkv2(wr[0], wr[1]); bw[t][1] = mkv2(wr[4], wr[5]); bw[t][2] = mkv2(wr[8], wr[9]); }
        else           { bw[t][0] = mkv2(wr[2], wr[3]); bw[t][1] = mkv2(wr[6], wr[7]); bw[t][2] = mkv2(0.f, 0.f); }
    }
}

// ---------------------------------------------------------------------------
// Kernel 2: BN stats of pre-BN LSE encoder output over (B,N,K) per channel.
// One wave per point (8 waves/block).
// ---------------------------------------------------------------------------
__global__ __launch_bounds__(256) void lse_stats_kernel(const float* __restrict__ pos,
                                                        const int* __restrict__ knn,
                                                        const float* __restrict__ lw,
                                                        const float* __restrict__ lb,
                                                        float* __restrict__ grp) {
    __shared__ float sSum[32], sSsq[32];
    const int w = threadIdx.x >> 5, lane = threadIdx.x & 31;
    if (threadIdx.x < 32) { sSum[threadIdx.x] = 0.f; sSsq[threadIdx.x] = 0.f; }
    __syncthreads();

    const int p = blockIdx.x * 8 + w;
    const int b = p / NN, n = p % NN;
    const int m = lane & 15;

    const EncFrags ef = build_enc_frags(pos, knn + (size_t)p * KK, b, n, lane);
    v2f bw[2][3];
    lse_wfrags(lw, lane, bw);

    v8f acc0 = zero8(), acc1 = zero8();
    #pragma unroll
    for (int s = 0; s < 3; ++s) acc0 = wmma4(ef.a[s], bw[0][s], acc0);
    #pragma unroll
    for (int s = 0; s < 3; ++s) acc1 = wmma4(ef.a[s], bw[1][s], acc1);

    #pragma unroll
    for (int t = 0; t < 2; ++t) {
        const int d = t * 16 + m;
        const float bi = lb[d];
        float s1 = 0.f, s2 = 0.f;
        #pragma unroll
        for (int v = 0; v < 8; ++v) {
            const float val = (t ? acc1[v] : acc0[v]) + bi;
            s1 += val; s2 += val * val;
        }
        s1 += __shfl_xor(s1, 16);
        s2 += __shfl_xor(s2, 16);
        if ((lane >> 4) == 0) { atomicAdd(&sSum[d], s1); atomicAdd(&sSsq[d], s2); }
    }
    __syncthreads();
    if (threadIdx.x < 32) {
        atomicAdd(&grp[threadIdx.x], sSum[threadIdx.x]);
        atomicAdd(&grp[32 + threadIdx.x], sSsq[threadIdx.x]);
    }
}

// ---------------------------------------------------------------------------
// Kernel 3: finalize BN stats (mean, invstd) for one group.
// ---------------------------------------------------------------------------
__global__ void finalize_stats(float* __restrict__ grp, int C, float cnt) {
    const int c = threadIdx.x;
    if (c >= C) return;
    const float mu = grp[c] / cnt;
    const float var = grp[C + c] / cnt - mu * mu;
    grp[2 * C + c] = mu;
    grp[3 * C + c] = rsqrtf(var + 1e-6f);
}

// ---------------------------------------------------------------------------
// Kernel 4: mlp1  x1 = leaky_relu(feats0 @ w1^T + b1, 0.2), (B,N,32)
// ---------------------------------------------------------------------------
__global__ void mlp1_kernel(const float* __restrict__ features,
                            const float* __restrict__ w1,
                            const float* __restrict__ b1,
                            float* __restrict__ x1) {
    const int tid = blockIdx.x * blockDim.x + threadIdx.x;
    if (tid >= BB * NN * HH) return;
    const int c = tid & 31;
    const int p = tid >> 5;
    const int b = p / NN, n = p % NN;
    float acc = b1[c];
    const float* wr = w1 + (size_t)c * D_IN;
    for (int i = 0; i < D_IN; ++i)
        acc += features[((size_t)b * D_IN + i) * NN + n] * wr[i];
    x1[tid] = acc >= 0.f ? acc : 0.2f * acc;
}

// ---------------------------------------------------------------------------
// Kernel 5: fused LSE (BN+ReLU+concat) + AttentivePooling. One wave per point.
//   - x[16][64] staged in LDS (enc_bn cols 0..31, feats cols 32..63)
//   - scores_pre = x @ pws^T via 64 WMMAs, softmax over K (rows) per channel
//   - pooled f -> pre-BN MLP output stored to gout; BN stats accumulated.
// fstats==nullptr -> feats used raw (stage 1); else BN+ReLU applied (stage 2).
// ---------------------------------------------------------------------------
__global__ __launch_bounds__(256) void lse_pool_kernel(
    const float* __restrict__ pos, const int* __restrict__ knn,
    const float* __restrict__ lw, const float* __restrict__ lb,
    const float* __restrict__ lstats, const float* __restrict__ lg,
    const float* __restrict__ lbe,
    const float* __restrict__ feats, const float* __restrict__ fstats,
    const float* __restrict__ fg, const float* __restrict__ fbe,
    const float* __restrict__ pws, const float* __restrict__ pw,
    const float* __restrict__ pbias,
    float* __restrict__ gout, float* __restrict__ pstats, int Cout) {
    __shared__ float xs[8][16][68];
    __shared__ float fvec[8][64];
    __shared__ float pSum[64], pSsq[64];

    const int w = threadIdx.x >> 5, lane = threadIdx.x & 31;
    const int m = lane & 15, half = lane >> 4;
    if (threadIdx.x < 64) { pSum[threadIdx.x] = 0.f; pSsq[threadIdx.x] = 0.f; }
    __syncthreads();

    const int p = blockIdx.x * 8 + w;
    const int b = p / NN, n = p % NN;

    // --- LSE encoder pre-BN via WMMA ---
    const EncFrags ef = build_enc_frags(pos, knn + (size_t)p * KK, b, n, lane);
    v2f bw[2][3];
    lse_wfrags(lw, lane, bw);
    v8f acc0 = zero8(), acc1 = zero8();
    #pragma unroll
    for (int s = 0; s < 3; ++s) acc0 = wmma4(ef.a[s], bw[0][s], acc0);
    #pragma unroll
    for (int s = 0; s < 3; ++s) acc1 = wmma4(ef.a[s], bw[1][s], acc1);

    const float* lmean = lstats + 64;   // C=32: mean at 2C
    const float* linv  = lstats + 96;   //       invstd at 3C
    #pragma unroll
    for (int t = 0; t < 2; ++t) {
        const int d = t * 16 + m;
        const float bi = lb[d], mu = lmean[d], iv = linv[d], gg = lg[d], be = lbe[d];
        #pragma unroll
        for (int v = 0; v < 8; ++v) {
            float val = ((t ? acc1[v] : acc0[v]) + bi - mu) * iv * gg + be;
            xs[w][v + 8 * half][d] = fmaxf(val, 0.f);
        }
    }
    // feats columns 32..63 broadcast over the K rows
    {
        const int c = lane;
        float fv = feats[(size_t)p * 32 + c];
        if (fstats) {
            fv = (fv - fstats[64 + c]) * fstats[96 + c] * fg[c] + fbe[c];
            fv = fmaxf(fv, 0.f);
        }
        #pragma unroll
        for (int r = 0; r < 16; ++r) xs[w][r][32 + c] = fv;
    }
    __syncthreads();

    // --- A fragments of x (shared across the 4 output tiles) ---
    v2f ax[16];
    #pragma unroll
    for (int s = 0; s < 16; ++s) {
        const int e = 4 * s + 2 * half;
        ax[s] = mkv2(xs[w][m][e], xs[w][m][e + 1]);
    }

    // --- scores = softmax_K(x @ pws^T); f = sum_K scores * x ---
    for (int t = 0; t < 4; ++t) {
        const int d = t * 16 + m;
        const float* wr = pws + (size_t)d * 64;
        v8f sp = zero8();
        #pragma unroll
        for (int s = 0; s < 16; ++s) {
            const v2f bb = mkv2(wr[4 * s + 2 * half], wr[4 * s + 2 * half + 1]);
            sp = wmma4(ax[s], bb, sp);
        }
        float mx = sp[0];
        #pragma unroll
        for (int v = 1; v < 8; ++v) mx = fmaxf(mx, sp[v]);
        mx = fmaxf(mx, __shfl_xor(mx, 16));
        float es[8], ssum = 0.f;
        #pragma unroll
        for (int v = 0; v < 8; ++v) { es[v] = __expf(sp[v] - mx); ssum += es[v]; }
        ssum += __shfl_xor(ssum, 16);
        const float inv = 1.f / ssum;
        float fpart = 0.f;
        #pragma unroll
        for (int v = 0; v < 8; ++v) fpart += es[v] * inv * xs[w][v + 8 * half][d];
        fpart += __shfl_xor(fpart, 16);
        if (half == 0) fvec[w][d] = fpart;
    }
    __syncthreads();

    // --- pooled MLP (pre-BN) + stats ---
    for (int j = lane; j < Cout; j += 32) {
        float acc = pbias[j];
        const float* wj = pw + (size_t)j * 64;
        for (int dd = 0; dd < 64; ++dd) acc += fvec[w][dd] * wj[dd];
        gout[(size_t)p * Cout + j] = acc;
        atomicAdd(&pSum[j], acc);
        atomicAdd(&pSsq[j], acc * acc);
    }
    __syncthreads();
    if (threadIdx.x < Cout) {
        atomicAdd(&pstats[threadIdx.x], pSum[threadIdx.x]);
        atomicAdd(&pstats[Cout + threadIdx.x], pSsq[threadIdx.x]);
    }
}

// ---------------------------------------------------------------------------
// Kernel 6: shortcut-BN stats over (B,N) per 128 channels.
// ---------------------------------------------------------------------------
__global__ __launch_bounds__(128) void sc_stats_kernel(const float* __restrict__ features,
                                                       const float* __restrict__ scw,
                                                       const float* __restrict__ scb,
                                                       float* __restrict__ grp) {
    const int j = threadIdx.x;
    const int p0 = blockIdx.x * 32;
    const float* wj = scw + (size_t)j * D_IN;
    const float bj = scb[j];
    float s1 = 0.f, s2 = 0.f;
    for (int t = 0; t < 32; ++t) {
        const int p = p0 + t;
        const int b = p / NN, n = p % NN;
        float acc = bj;
        for (int i = 0; i < D_IN; ++i)
            acc += features[((size_t)b * D_IN + i) * NN + n] * wj[i];
        s1 += acc; s2 += acc * acc;
    }
    atomicAdd(&grp[j], s1);
    atomicAdd(&grp[128 + j], s2);
}

// ---------------------------------------------------------------------------
// Kernel 7: out = leaky( mlp2(relu(bn(g2))) + bn(shortcut), 0.01 ), (B,128,N,1)
// ---------------------------------------------------------------------------
__global__ __launch_bounds__(128) void final_kernel(
    const float* __restrict__ features, const float* __restrict__ g2,
    const float* __restrict__ p2stats, const float* __restrict__ p2g,
    const float* __restrict__ p2be,
    const float* __restrict__ w2, const float* __restrict__ b2,
    const float* __restrict__ scw, const float* __restrict__ scb,
    const float* __restrict__ scstats, const float* __restrict__ scg,
    const float* __restrict__ scbe, float* __restrict__ out) {
    __shared__ float xp[64];
    const int p = blockIdx.x;
    const int b = p / NN, n = p % NN;
    const int j = threadIdx.x;
    if (j < 64) {
        float v = g2[(size_t)p * 64 + j];
        v = (v - p2stats[128 + j]) * p2stats[192 + j] * p2g[j] + p2be[j];
        xp[j] = fmaxf(v, 0.f);
    }
    __syncthreads();
    float acc = b2[j];
    const float* wr = w2 + (size_t)j * 64;
    for (int d = 0; d < 64; ++d) acc += xp[d] * wr[d];
    float sc = scb[j];
    const float* wj = scw + (size_t)j * D_IN;
    for (int i = 0; i < D_IN; ++i)
        sc += features[((size_t)b * D_IN + i) * NN + n] * wj[i];
    sc = (sc - scstats[256 + j]) * scstats[384 + j] * scg[j] + scbe[j];
    const float o = acc + sc;
    out[((size_t)b * 128 + j) * NN + n] = o >= 0.f ? o : 0.01f * o;
}

// ---------------------------------------------------------------------------
extern "C" void kernel_launch(void* const* d_in, const int* in_sizes, int n_in,
                              void* d_out, int out_size, void* d_ws, size_t ws_size,
                              hipStream_t stream) {
    (void)in_sizes; (void)n_in; (void)out_size; (void)ws_size;
    const float* pos      = (const float*)d_in[0];
    const float* features = (const float*)d_in[1];
    const float* w1      = (const float*)d_in[2];
    const float* b1      = (const float*)d_in[3];
    const float* lse1_w  = (const float*)d_in[4];
    const float* lse1_b  = (const float*)d_in[5];
    const float* lse1_g  = (const float*)d_in[6];
    const float* lse1_be = (const float*)d_in[7];
    const float* p1_ws   = (const float*)d_in[8];
    const float* p1_w    = (const float*)d_in[9];
    const float* p1_b    = (const float*)d_in[10];
    const float* p1_g    = (const float*)d_in[11];
    const float* p1_be   = (const float*)d_in[12];
    const float* lse2_w  = (const float*)d_in[13];
    const float* lse2_b  = (const float*)d_in[14];
    const float* lse2_g  = (const float*)d_in[15];
    const float* lse2_be = (const float*)d_in[16];
    const float* p2_ws   = (const float*)d_in[17];
    const float* p2_w    = (const float*)d_in[18];
    const float* p2_b    = (const float*)d_in[19];
    const float* p2_g    = (const float*)d_in[20];
    const float* p2_be   = (const float*)d_in[21];
    const float* w2      = (const float*)d_in[22];
    const float* b2      = (const float*)d_in[23];
    const float* sc_w    = (const float*)d_in[24];
    const float* sc_b    = (const float*)d_in[25];
    const float* sc_g    = (const float*)d_in[26];
    const float* sc_be   = (const float*)d_in[27];

    char* ws = (char*)d_ws;
    int*   idx   = (int*)ws;
    float* x1    = (float*)(ws + (size_t)BB * NN * KK * sizeof(int));
    float* g1    = x1 + (size_t)BB * NN * HH;
    float* g2    = g1 + (size_t)BB * NN * HH;
    float* stats = g2 + (size_t)BB * NN * D_OUT;

    (void)hipMemsetAsync(stats, 0, STATS_FLOATS * sizeof(float), stream);

    knn_kernel<<<BB * (NN / 128), 256, 0, stream>>>(pos, idx);
    mlp1_kernel<<<(BB * NN * HH + 255) / 256, 256, 0, stream>>>(features, w1, b1, x1);

    // stage 1: LSE1 + pool1
    lse_stats_kernel<<<BB * NN / 8, 256, 0, stream>>>(pos, idx, lse1_w, lse1_b,
                                                      stats + GRP_LSE1);
    finalize_stats<<<1, 128, 0, stream>>>(stats + GRP_LSE1, 32, (float)(BB * NN * KK));
    lse_pool_kernel<<<BB * NN / 8, 256, 0, stream>>>(
        pos, idx, lse1_w, lse1_b, stats + GRP_LSE1, lse1_g, lse1_be,
        x1, nullptr, nullptr, nullptr,
        p1_ws, p1_w, p1_b, g1, stats + GRP_P1, 32);
    finalize_stats<<<1, 128, 0, stream>>>(stats + GRP_P1, 32, (float)(BB * NN));

    // stage 2: LSE2 + pool2 (feats = relu(bn(g1)))
    lse_stats_kernel<<<BB * NN / 8, 256, 0, stream>>>(pos, idx, lse2_w, lse2_b,
                                                      stats + GRP_LSE2);
    finalize_stats<<<1, 128, 0, stream>>>(stats + GRP_LSE2, 32, (float)(BB * NN * KK));
    lse_pool_kernel<<<BB * NN / 8, 256, 0, stream>>>(
        pos, idx, lse2_w, lse2_b, stats + GRP_LSE2, lse2_g, lse2_be,
        g1, stats + GRP_P1, p1_g, p1_be,
        p2_ws, p2_w, p2_b, g2, stats + GRP_P2, 64);
    finalize_stats<<<1, 128, 0, stream>>>(stats + GRP_P2, 64, (float)(BB * NN));

    // shortcut BN stats + final fuse
    sc_stats_kernel<<<BB * NN / 32, 128, 0, stream>>>(features, sc_w, sc_b,
                                                      stats + GRP_SC);
    finalize_stats<<<1, 128, 0, stream>>>(stats + GRP_SC, 128, (float)(BB * NN));
    final_kernel<<<BB * NN, 128, 0, stream>>>(features, g2, stats + GRP_P2, p2_g, p2_be,
                                              w2, b2, sc_w, sc_b, stats + GRP_SC,
                                              sc_g, sc_be, (float*)d_out);
}